// UniformNeighborSampler_16492674417064
// MI455X (gfx1250) — compile-verified
//
#include <hip/hip_runtime.h>
#include <hip/hip_bf16.h>
#include <cmath>
#include <cstdint>

// ---------------------------------------------------------------------------
// UniformNeighborSampler top-k:
//   adj = adj_info[ids]            (BATCH x 128, int32, contiguous rows)
//   w   = prob_matrix[ids, adj]    (BATCH x 128, scattered fp32 gathers)
//   out = adj[ top_k(w, 25) ]      (descending value, ties -> smaller index)
//
// Strategy: one wave32 per row. Adjacency rows (512B) are staged into LDS
// with the CDNA5 async copy engine (global_load_async_to_lds_b128) and
// double-buffered across the wave's row loop so the next row's adjacency
// fetch overlaps the current row's scattered gathers + top-k reduction.
// Top-k = 25 rounds of wave-wide argmax via xor-butterfly shuffles with
// (value desc, position asc) ordering == jax.lax.top_k tie-breaking.
// ---------------------------------------------------------------------------

#define WAVES_PER_BLOCK 8
#define DEG 128          // fast path specialized for MAX_DEGREE == 128

__device__ __forceinline__ void async_row_b128(const void* gaddr, unsigned lds_addr) {
  // CDNA5 async copy: 16B per lane, 32 lanes -> 512B row into LDS.
  // VDST operand carries the per-lane LDS byte address; tracked by ASYNCcnt.
  asm volatile("global_load_async_to_lds_b128 %0, %1, off"
               :
               : "v"(lds_addr), "v"(gaddr)
               : "memory");
}
__device__ __forceinline__ void wait_async_le(int) {}
__device__ __forceinline__ void wait_async_le1() {
  asm volatile("s_wait_asynccnt 1" ::: "memory");
}
__device__ __forceinline__ void wait_async_le0() {
  asm volatile("s_wait_asynccnt 0" ::: "memory");
}

__global__ __launch_bounds__(256) void topk_neighbor_fast(
    const int* __restrict__ ids, const int* __restrict__ adj,
    const float* __restrict__ prob, int* __restrict__ out,
    int batch, int n_nodes, int k, int wave_stride) {
  __shared__ int adjbuf[WAVES_PER_BLOCK * 2 * DEG];

  const int lane = threadIdx.x & 31;
  const int wv   = threadIdx.x >> 5;
  const int gw   = blockIdx.x * WAVES_PER_BLOCK + wv;   // global wave id
  if (gw >= batch) return;  // no block-wide barriers used -> safe early exit

  int* buf0 = &adjbuf[(wv * 2 + 0) * DEG];
  int* buf1 = &adjbuf[(wv * 2 + 1) * DEG];
  // Low 32 bits of a generic pointer to LDS == wave-relative LDS byte address.
  const unsigned lds0 =
      (unsigned)(uint64_t)(void*)buf0 + (unsigned)lane * 16u;
  const unsigned lds1 =
      (unsigned)(uint64_t)(void*)buf1 + (unsigned)lane * 16u;

  const float NEG = -__builtin_huge_valf();

  // Prologue: kick off async copy of the first row's adjacency list.
  int curId = ids[gw];
  async_row_b128(adj + (size_t)curId * DEG + lane * 4, lds0);

  int parity = 0;
  for (int row = gw; row < batch; row += wave_stride) {
    const int nrow = row + wave_stride;
    int nextId = 0;
    if (nrow < batch) {
      // Prefetch next row's adjacency into the other buffer, then wait only
      // for the *older* copy (async loads complete in order).
      nextId = ids[nrow];
      async_row_b128(adj + (size_t)nextId * DEG + lane * 4,
                     parity ? lds0 : lds1);
      wait_async_le1();
    } else {
      wait_async_le0();
    }

    const int* mybuf = parity ? buf1 : buf0;
    const int4 a = *reinterpret_cast<const int4*>(mybuf + lane * 4);

    // 4 independent scattered gathers per lane -> deep MLP of outstanding
    // loads across 2048 waves (latency-bound phase).
    const float* prow = prob + (size_t)curId * (size_t)n_nodes;
    float v0 = prow[a.x];
    float v1 = prow[a.y];
    float v2 = prow[a.z];
    float v3 = prow[a.w];
    const int nb0 = a.x, nb1 = a.y, nb2 = a.z, nb3 = a.w;

    const int p0 = lane * 4, p1 = p0 + 1, p2 = p0 + 2, p3 = p0 + 3;

    int myres = 0;
    for (int it = 0; it < k; ++it) {
      // Lane-local argmax (strict > keeps the smallest position on ties).
      float bv = v0; int bp = p0;
      if (v1 > bv) { bv = v1; bp = p1; }
      if (v2 > bv) { bv = v2; bp = p2; }
      if (v3 > bv) { bv = v3; bp = p3; }
      // Wave-wide xor-butterfly all-reduce argmax with top_k tie-breaking.
#pragma unroll
      for (int off = 16; off > 0; off >>= 1) {
        const float ov = __shfl_xor(bv, off, 32);
        const int   op = __shfl_xor(bp, off, 32);
        if (ov > bv || (ov == bv && op < bp)) { bv = ov; bp = op; }
      }
      // All lanes agree on winner position bp. Fetch winner's neighbor id.
      const int slot = bp & 3;
      int cand = nb0;
      cand = (slot == 1) ? nb1 : cand;
      cand = (slot == 2) ? nb2 : cand;
      cand = (slot == 3) ? nb3 : cand;
      const int win = __shfl(cand, bp >> 2, 32);
      if (lane == it) myres = win;   // lane `it` keeps the it-th result
      // Knock out the winner (branch-free).
      v0 = (bp == p0) ? NEG : v0;
      v1 = (bp == p1) ? NEG : v1;
      v2 = (bp == p2) ? NEG : v2;
      v3 = (bp == p3) ? NEG : v3;
    }
    // Coalesced output: lanes 0..k-1 each store one int32.
    if (lane < k) out[(size_t)row * k + lane] = myres;

    curId = nextId;
    parity ^= 1;
  }
}

// Generic fallback (any degree / k): one thread per row, k passes of
// threshold-filtered argmax; exact top_k ordering, O(k*deg), no scratch.
__global__ void topk_neighbor_generic(
    const int* __restrict__ ids, const int* __restrict__ adj,
    const float* __restrict__ prob, int* __restrict__ out,
    int batch, int n_nodes, int deg, int k) {
  const int r = blockIdx.x * blockDim.x + threadIdx.x;
  if (r >= batch) return;
  const int id = ids[r];
  const int*   arow = adj  + (size_t)id * deg;
  const float* prow = prob + (size_t)id * n_nodes;
  float lastV = __builtin_huge_valf();
  int   lastP = -1;
  for (int it = 0; it < k; ++it) {
    float bv = -__builtin_huge_valf();
    int bp = deg, bn = 0;
    for (int j = 0; j < deg; ++j) {
      const int c = arow[j];
      const float w = prow[c];
      const bool cand   = (w < lastV) || (w == lastV && j > lastP);
      const bool better = (w > bv)    || (w == bv    && j < bp);
      if (cand && better) { bv = w; bp = j; bn = c; }
    }
    out[(size_t)r * k + it] = bn;
    lastV = bv; lastP = bp;
  }
}

extern "C" void kernel_launch(void* const* d_in, const int* in_sizes, int n_in,
                              void* d_out, int out_size, void* d_ws, size_t ws_size,
                              hipStream_t stream) {
  const int*   ids  = (const int*)d_in[0];
  const int*   adj  = (const int*)d_in[1];
  const float* prob = (const float*)d_in[2];
  int* out = (int*)d_out;   // reference output dtype is int32

  const int batch = in_sizes[0];
  const int n_nodes = (int)(sqrt((double)in_sizes[2]) + 0.5);
  const int deg = (n_nodes > 0) ? in_sizes[1] / n_nodes : 0;
  const int k = (batch > 0) ? out_size / batch : 0;
  if (batch <= 0 || k <= 0) return;

  if (deg == DEG && k <= 32) {
    // ~2 rows per wave so the async adjacency double-buffer has overlap,
    // while keeping ~2048 waves in flight for gather latency hiding.
    int waves_needed = (batch + 1) / 2;
    int blocks = (waves_needed + WAVES_PER_BLOCK - 1) / WAVES_PER_BLOCK;
    if (blocks < 1) blocks = 1;
    const int wave_stride = blocks * WAVES_PER_BLOCK;
    topk_neighbor_fast<<<blocks, WAVES_PER_BLOCK * 32, 0, stream>>>(
        ids, adj, prob, out, batch, n_nodes, k, wave_stride);
  } else {
    const int threads = 256;
    const int blocks = (batch + threads - 1) / threads;
    topk_neighbor_generic<<<blocks, threads, 0, stream>>>(
        ids, adj, prob, out, batch, n_nodes, deg, k);
  }
}